// LIF_16312285791153
// MI455X (gfx1250) — compile-verified
//
#include <hip/hip_runtime.h>

// ---------------------------------------------------------------------------
// LIF neuron scan for MI455X (gfx1250).
// Memory-bound streaming recurrence: 210 MB total traffic => ~9us at 23.3 TB/s.
// CDNA5 path used: async global->LDS copy engine (ASYNCcnt) as a double-
// buffered prefetch pipeline, waited with s_wait_asynccnt. wave32 sizing.
// ---------------------------------------------------------------------------

#define LSEQ    400   // timesteps (compile-time, from reference)
#define T_CHUNK 20    // timesteps staged per async tile
#define N_TILES (LSEQ / T_CHUNK)
#define BLOCK   256   // 8 waves of 32

typedef __attribute__((address_space(1))) int g_int;  // global (device) int
typedef __attribute__((address_space(3))) int l_int;  // LDS int

// ---- async global -> LDS copy (gfx1250), builtin if present, else asm ------
__device__ __forceinline__ void async_load_f32(const float* g, float* l) {
#if __has_builtin(__builtin_amdgcn_global_load_async_to_lds_b32)
  __builtin_amdgcn_global_load_async_to_lds_b32(
      (g_int*)(void*)g, (l_int*)l, /*offset=*/0, /*cpol=*/0);
#else
  unsigned loff = (unsigned)(__UINTPTR_TYPE__)(l_int*)l;
  asm volatile("global_load_async_to_lds_b32 %0, %1, off"
               :: "v"(loff), "v"(g)
               : "memory");
#endif
}

#if __has_builtin(__builtin_amdgcn_s_wait_asynccnt)
#define WAIT_ASYNCCNT(n) __builtin_amdgcn_s_wait_asynccnt(n)
#else
#define WAIT_ASYNCCNT(n) asm volatile("s_wait_asynccnt %0" :: "n"(n) : "memory")
#endif

__global__ __launch_bounds__(BLOCK) void lif_scan_kernel(
    const float* __restrict__ I, float* __restrict__ spikes,
    float* __restrict__ hardlat, float* __restrict__ softlat, int B) {
  // Time-major LDS tile: lds reads in the hot loop are bank-conflict-free
  // (consecutive lanes -> consecutive banks). 2*20*256*4 = 40 KB.
  __shared__ float tile[2][T_CHUNK][BLOCK];

  const int tid = threadIdx.x;
  const int n = blockIdx.x * BLOCK + tid;
  const bool active = (n < B);
  const int row = active ? n : (B - 1);  // clamp, keep wave-uniform issue

  const float* rowBase = I + (size_t)row * LSEQ;
  float* outRow = spikes + (size_t)row * LSEQ;

  // Prologue: stage tile 0 into buffer 0 (20 async ops, ASYNCcnt += 20).
#pragma unroll
  for (int t = 0; t < T_CHUNK; ++t)
    async_load_f32(rowBase + t, &tile[0][t][tid]);

  // Neuron state in registers.
  float v = 0.0f;
  float theta = 1.0f;   // TH
  int refc = 0;
  float sum_s = 0.0f;
  float sum_st = 0.0f;
  int first = LSEQ;     // 400 == "no spike" sentinel (matches reference)

  for (int k = 0; k < N_TILES; ++k) {
    const int buf = k & 1;

    if (k + 1 < N_TILES) {
      // Prefetch next tile into the other buffer (ASYNCcnt peaks at 40 < 63),
      // then wait until only those 20 remain outstanding -> tile k is ready
      // (async loads complete in order).
      const float* gb = rowBase + (k + 1) * T_CHUNK;
#pragma unroll
      for (int t = 0; t < T_CHUNK; ++t)
        async_load_f32(gb + t, &tile[buf ^ 1][t][tid]);
      WAIT_ASYNCCNT(T_CHUNK);
    } else {
      WAIT_ASYNCCNT(0);
    }
    // Compiler barrier: keep the ds_load_b32 reads below the wait.
    asm volatile("" ::: "memory");

#pragma unroll
    for (int t4 = 0; t4 < T_CHUNK; t4 += 4) {
      float ss[4];
#pragma unroll
      for (int j = 0; j < 4; ++j) {
        const int t = t4 + j;
        const float It = tile[buf][t][tid];

        // v = v + (-v / TAU + I_t)
        v = v + (-v / 20.0f + It);
        const bool can = (refc == 0);
        const float x = v - theta;
        // surrogate sigmoid(K*x) gated by refractory
        const float sig = 1.0f / (1.0f + __expf(-10.0f * x));
        const float s_soft = can ? sig : 0.0f;
        const float s_hard = (x >= 0.0f && can) ? 1.0f : 0.0f;
        // straight-through forward value, same rounding as reference
        const float s = (s_hard - s_soft) + s_soft;

        refc = (s_hard > 0.0f) ? 5 : (refc > 0 ? refc - 1 : 0);
        v = v * (1.0f - s);
        theta = theta - (theta - 1.0f) / 50.0f + 0.5f * s_hard;

        const int tg = k * T_CHUNK + t;
        if (s_hard > 0.0f && first == LSEQ) first = tg;
        sum_s += s;
        sum_st += s * (float)tg;
        ss[j] = s;
      }
      if (active) {
        // 16B-aligned: row stride 1600B, offset multiple of 16B.
        *(float4*)(outRow + k * T_CHUNK + t4) =
            make_float4(ss[0], ss[1], ss[2], ss[3]);
      }
    }
  }

  if (active) {
    hardlat[n] = (float)first;                 // int64 latency, exact in f32
    softlat[n] = sum_st / (sum_s + 1e-6f);     // sum(s*t) / (sum(s)+eps)
  }
}

extern "C" void kernel_launch(void* const* d_in, const int* in_sizes, int n_in,
                              void* d_out, int out_size, void* d_ws,
                              size_t ws_size, hipStream_t stream) {
  (void)n_in; (void)out_size; (void)d_ws; (void)ws_size;
  const float* I = (const float*)d_in[0];
  const int B = in_sizes[0] / LSEQ;  // 65536

  float* out = (float*)d_out;
  float* spikes = out;                         // [B, 400]
  float* hardlat = out + (size_t)B * LSEQ;     // [B]
  float* softlat = hardlat + B;                // [B]

  dim3 grid((B + BLOCK - 1) / BLOCK);
  dim3 block(BLOCK);
  hipLaunchKernelGGL(lif_scan_kernel, grid, block, 0, stream,
                     I, spikes, hardlat, softlat, B);
}